// NMS_3762391351432
// MI455X (gfx1250) — compile-verified
//
#include <hip/hip_runtime.h>

#define NCOLS 576
#define MROWS 144
#define ROW_DEG 15
#define BATCH 256
#define ITERS 3
#define MAXD 24   // max column degree we store (true max ~12 for this H)

// d_ws layout (ints):
//   [0,                2160)                 rowCols[m*15+k]   : column index of slot k of row m
//   [2160,             2160+576)             colDeg[n]
//   [2736,             2736+576*MAXD)        colEntry[n*MAXD+j]: (m<<4)|k of entries hitting column n

__global__ __launch_bounds__(576) void ldpc_build_struct(
    const float* __restrict__ H,
    int* __restrict__ rowCols,
    int* __restrict__ colDeg,
    int* __restrict__ colEntry) {
  const int tid = threadIdx.x;
  // Phase A: CSR — per-row column indices, in ascending column order.
  if (tid < MROWS) {
    int k = 0;
    const float* hrow = H + tid * NCOLS;
    for (int n = 0; n < NCOLS; ++n) {
      if (hrow[n] == 1.0f) {
        if (k < ROW_DEG) rowCols[tid * ROW_DEG + k] = n;
        ++k;
      }
    }
  }
  __syncthreads();
  // Phase B: CSC — per-column incident (row,slot) list, deterministic order.
  if (tid < NCOLS) {
    int deg = 0;
    for (int m = 0; m < MROWS; ++m) {
      for (int k = 0; k < ROW_DEG; ++k) {
        if (rowCols[m * ROW_DEG + k] == tid) {
          if (deg < MAXD) colEntry[tid * MAXD + deg] = (m << 4) | k;
          ++deg;
        }
      }
    }
    colDeg[tid] = (deg < MAXD) ? deg : MAXD;
  }
}

__global__ __launch_bounds__(192) void ldpc_minsum(
    const float* __restrict__ r,
    const float* __restrict__ alpha,
    const float* __restrict__ beta,
    const int* __restrict__ rowCols,
    const int* __restrict__ colDeg,
    const int* __restrict__ colEntry,
    float* __restrict__ out) {
  __shared__ __align__(16) float s_r[NCOLS];
  __shared__ __align__(16) float s_sum[NCOLS];
  __shared__ float s_E[MROWS * ROW_DEG];
  __shared__ float s_M[MROWS * ROW_DEG];
  __shared__ int   s_cols[MROWS * ROW_DEG];

  const int b    = blockIdx.x;
  const int tid  = threadIdx.x;
  const int nthr = blockDim.x;  // 192

  // ---- Stage r[b,:] into LDS via CDNA5 async global->LDS DMA (B128) ----
  {
    const float* gsrc = r + (size_t)b * NCOLS;
    if (tid < NCOLS / 4) {
      // Low 32 bits of a generic LDS pointer are the LDS (as3) byte address.
      unsigned lds_addr = (unsigned)(unsigned long long)(&s_r[tid * 4]);
      const float* ga = gsrc + tid * 4;
      asm volatile("global_load_async_to_lds_b128 %0, %1, off"
                   :: "v"(lds_addr), "v"(ga)
                   : "memory");
    }
    asm volatile("s_wait_asynccnt 0x0" ::: "memory");
  }
  for (int i = tid; i < MROWS * ROW_DEG; i += nthr)
    s_cols[i] = rowCols[i];
  __syncthreads();

  // M = H * r at support positions
  for (int i = tid; i < MROWS * ROW_DEG; i += nthr)
    s_M[i] = s_r[s_cols[i]];
  __syncthreads();

  for (int it = 0; it < ITERS; ++it) {
    const float a_ = alpha[it];
    const float b_ = beta[it];

    // ---- Check-node (row) phase: min1/min2/argmin + sign product ----
    if (tid < MROWS) {
      float v[ROW_DEG];
      float min1 = __builtin_inff(), min2 = __builtin_inff();
      int   argc_ = 0x7FFFFFFF, argk = -1;
      float sprod = 1.0f;
      const int base = tid * ROW_DEG;
      for (int k = 0; k < ROW_DEG; ++k) {
        const float x = s_M[base + k];
        v[k] = x;
        const float sg = (x > 0.0f) ? 1.0f : ((x < 0.0f) ? -1.0f : 0.0f);
        sprod *= sg;
        const float av  = fabsf(x);
        const int   col = s_cols[base + k];
        // JAX top_k tie-break: lowest column index wins the argmin slot.
        if (av < min1 || (av == min1 && col < argc_)) {
          min2 = min1; min1 = av; argc_ = col; argk = k;
        } else if (av < min2) {
          min2 = av;
        }
      }
      for (int k = 0; k < ROW_DEG; ++k) {
        const float x  = v[k];
        const float sg = (x > 0.0f) ? 1.0f : ((x < 0.0f) ? -1.0f : 0.0f);
        const float eabs = (k == argk) ? min2 : min1;
        float m = eabs - b_;
        m = (m > 0.0f) ? m : 0.0f;
        s_E[base + k] = a_ * sprod * sg * m;
      }
    }
    __syncthreads();

    // ---- Column-sum phase: deterministic CSC gather (no atomics).
    //      On the final iteration, fold in r so s_sum holds the output. ----
    const float init = (it == ITERS - 1) ? 1.0f : 0.0f;
    for (int n = tid; n < NCOLS; n += nthr) {
      const int  deg = colDeg[n];
      const int* ce  = colEntry + n * MAXD;
      float s = init * s_r[n];
      for (int j = 0; j < deg; ++j) {
        const int e = ce[j];
        s += s_E[(e >> 4) * ROW_DEG + (e & 15)];
      }
      s_sum[n] = s;
    }
    __syncthreads();

    // ---- Variable-node update (not needed after the last iteration) ----
    if (it + 1 < ITERS) {
      for (int i = tid; i < MROWS * ROW_DEG; i += nthr) {
        const int col = s_cols[i];
        s_M[i] = s_r[col] + s_sum[col] - s_E[i];
      }
      __syncthreads();
    }
  }

  // ---- out[b,:] = s_sum via CDNA5 async LDS->global DMA (B128) ----
  if (tid < NCOLS / 4) {
    unsigned lds_addr = (unsigned)(unsigned long long)(&s_sum[tid * 4]);
    float* ga = out + (size_t)b * NCOLS + tid * 4;
    asm volatile("global_store_async_from_lds_b128 %0, %1, off"
                 :: "v"(ga), "v"(lds_addr)
                 : "memory");
  }
  asm volatile("s_wait_asynccnt 0x0" ::: "memory");
}

extern "C" void kernel_launch(void* const* d_in, const int* in_sizes, int n_in,
                              void* d_out, int out_size, void* d_ws, size_t ws_size,
                              hipStream_t stream) {
  const float* r     = (const float*)d_in[0];  // (256, 576)
  const float* H     = (const float*)d_in[1];  // (144, 576)
  const float* alpha = (const float*)d_in[2];  // (3,)
  const float* beta  = (const float*)d_in[3];  // (3,)
  float* out = (float*)d_out;                  // (256, 576)

  int* ws       = (int*)d_ws;
  int* rowCols  = ws;                                  // 2160
  int* colDeg   = ws + MROWS * ROW_DEG;                // 576
  int* colEntry = colDeg + NCOLS;                      // 576*MAXD

  ldpc_build_struct<<<1, 576, 0, stream>>>(H, rowCols, colDeg, colEntry);
  ldpc_minsum<<<BATCH, 192, 0, stream>>>(r, alpha, beta, rowCols, colDeg,
                                         colEntry, out);
}